// MRConv4d_54022098649946
// MI455X (gfx1250) — compile-verified
//
#include <hip/hip_runtime.h>
#include <hip/hip_bf16.h>

#define BB   16
#define CC   96
#define HH   112
#define WW   112
#define HW   (HH*WW)            // 12544
#define C2   192
#define OUTC 192
#define NPOS (BB*HW)            // 200704
#define NTILES (NPOS/16)        // 12544
#define EPSV 1e-5f

typedef __attribute__((ext_vector_type(2))) float v2f;
typedef __attribute__((ext_vector_type(8))) float v8f;

// ---------------------------------------------------------------------------
// Kernel 0: zero the stats accumulators (sum + sumsq, 384 floats)
// ---------------------------------------------------------------------------
__global__ void zero_stats_kernel(float* __restrict__ g) {
    int i = threadIdx.x;
    if (i < 2*OUTC) g[i] = 0.0f;
}

// ---------------------------------------------------------------------------
// Kernel 1: xj[b,c,h,w] = max(0, xs - minExclSelf(same-parity col),
//                                xs - minExclSelf(same-parity row))
// where xs[h][w] = x[(h+1)%112][(w+1)%112].  One block per (b,c) plane.
// Exactly reproduces the 110-roll max-scan via min/2nd-min + argmin
// (shift set {2,4,...,110} mod 112 == all same-parity indices except self).
// ---------------------------------------------------------------------------
__global__ void shiftmax_kernel(const float* __restrict__ x, float* __restrict__ xj) {
    __shared__ float pl[HW];                 // 50176 B  (xs plane)
    __shared__ float cm1[WW*2], cm2[WW*2];   // column mins per (w, parity)
    __shared__ int   ci1[WW*2];
    __shared__ float rm1[HH*2], rm2[HH*2];   // row mins per (h, parity)
    __shared__ int   ri1[HH*2];

    const int plane = blockIdx.x;            // b*96 + c
    const float* __restrict__ xp = x  + (size_t)plane * HW;
    float* __restrict__ xjp      = xj + (size_t)plane * HW;

    // load pre-shifted plane: pl[h][w] = x[(h+1)%H][(w+1)%W]
    for (int i = threadIdx.x; i < HW; i += blockDim.x) {
        int h = i / WW, w = i - h*WW;
        int hs = (h + 1 == HH) ? 0 : h + 1;
        int ws_ = (w + 1 == WW) ? 0 : w + 1;
        pl[i] = xp[hs*WW + ws_];
    }
    __syncthreads();

    if (threadIdx.x < WW*2) {                // column task: w = t>>1, parity = t&1
        int t = threadIdx.x, w = t >> 1, p = t & 1;
        float m1 = 3.4e38f, m2 = 3.4e38f; int i1 = -1;
        for (int h = p; h < HH; h += 2) {
            float v = pl[h*WW + w];
            if (v < m1) { m2 = m1; m1 = v; i1 = h; }
            else if (v < m2) { m2 = v; }
        }
        cm1[t] = m1; cm2[t] = m2; ci1[t] = i1;
    }
    if (threadIdx.x < HH*2) {                // row task: h = t>>1, parity = t&1
        int t = threadIdx.x, h = t >> 1, p = t & 1;
        float m1 = 3.4e38f, m2 = 3.4e38f; int i1 = -1;
        for (int w = p; w < WW; w += 2) {
            float v = pl[h*WW + w];
            if (v < m1) { m2 = m1; m1 = v; i1 = w; }
            else if (v < m2) { m2 = v; }
        }
        rm1[t] = m1; rm2[t] = m2; ri1[t] = i1;
    }
    __syncthreads();

    for (int i = threadIdx.x; i < HW; i += blockDim.x) {
        int h = i / WW, w = i - h*WW;
        float v = pl[i];
        int tc = (w << 1) | (h & 1);
        float mh = (h == ci1[tc]) ? cm2[tc] : cm1[tc];
        int tr = (h << 1) | (w & 1);
        float mw = (w == ri1[tr]) ? rm2[tr] : rm1[tr];
        float r = fmaxf(v - mh, v - mw);
        xjp[i] = fmaxf(r, 0.0f);
    }
}

// ---------------------------------------------------------------------------
// Kernel 2: 1x1 conv as f32 WMMA GEMM + bias, fused per-channel sum/sumsq.
// Each wave: 16 consecutive spatial positions x all 192 output channels.
// A (16x4 f32): lane L (m=L&15, kh=L>>4) holds cat[m][k0], cat[m][k0+1],
//               k0 = 4*kk + 2*kh.  B (4x16): lane L holds w[n=m][k0], w[n][k0+1].
// C/D (16x16 f32): reg i, lanes0-15 -> (M=i,N=L); lanes16-31 -> (M=i+8,N=L-16).
// K loop split at channel 96 (x half vs xj half) -> branch-free inner loops.
// ---------------------------------------------------------------------------
__global__ void __launch_bounds__(256)
gemm_wmma_kernel(const float* __restrict__ x, const float* __restrict__ xj,
                 const float* __restrict__ cw, const float* __restrict__ cb,
                 float* __restrict__ y, float* __restrict__ gstats) {
    __shared__ float lsum[OUTC];
    __shared__ float lsumsq[OUTC];
    for (int i = threadIdx.x; i < OUTC; i += blockDim.x) { lsum[i] = 0.f; lsumsq[i] = 0.f; }
    __syncthreads();

    const int wave = threadIdx.x >> 5;
    const int lane = threadIdx.x & 31;
    const int tile = blockIdx.x * (blockDim.x >> 5) + wave;   // 0..12543
    const int p0   = tile * 16;
    const int b    = p0 / HW;
    const int rem  = p0 - b * HW;
    const int h    = rem / WW;
    const int w0   = rem - h * WW;                            // 112 % 16 == 0: tile stays in one row
    const int m    = lane & 15;
    const int kh   = lane >> 4;                               // 0 or 1

    const size_t posoff = (size_t)h * WW + w0 + m;
    const float* __restrict__ xb_base = x  + (size_t)b * CC * HW + posoff + (size_t)kh * 2 * HW;
    const float* __restrict__ xj_base = xj + (size_t)b * CC * HW + posoff + (size_t)kh * 2 * HW;
    const float* __restrict__ wb     = cw + kh * 2;           // lane's K pair within conv_w row

    v8f acc[12] = {};

    // --- first half: channels 0..95 come from xb == original x ---
    for (int kk = 0; kk < 24; ++kk) {
        v2f a;
        a.x = xb_base[(size_t)(kk * 4) * HW];
        a.y = xb_base[(size_t)(kk * 4 + 1) * HW];
        #pragma unroll
        for (int nb = 0; nb < 12; ++nb) {
            const float* wp = wb + (nb * 16 + m) * C2 + kk * 4;   // 8B-aligned pair
            v2f bv; bv.x = wp[0]; bv.y = wp[1];
            acc[nb] = __builtin_amdgcn_wmma_f32_16x16x4_f32(
                false, a, false, bv, (short)0, acc[nb], false, false);
        }
    }
    // --- second half: channels 96..191 come from xj ---
    for (int kk = 0; kk < 24; ++kk) {
        v2f a;
        a.x = xj_base[(size_t)(kk * 4) * HW];
        a.y = xj_base[(size_t)(kk * 4 + 1) * HW];
        #pragma unroll
        for (int nb = 0; nb < 12; ++nb) {
            const float* wp = wb + (nb * 16 + m) * C2 + 96 + kk * 4;
            v2f bv; bv.x = wp[0]; bv.y = wp[1];
            acc[nb] = __builtin_amdgcn_wmma_f32_16x16x4_f32(
                false, a, false, bv, (short)0, acc[nb], false, false);
        }
    }

    // epilogue: bias, store y, accumulate per-channel sum / sumsq
    #pragma unroll
    for (int nb = 0; nb < 12; ++nb) {
        const int o = nb * 16 + m;
        const float bo = cb[o];
        float s = 0.f, s2 = 0.f;
        float* __restrict__ yp = y + (((size_t)b * OUTC + o) * HH + h) * WW + w0 + kh * 8;
        #pragma unroll
        for (int i = 0; i < 8; ++i) {
            float v = acc[nb][i] + bo;
            s += v; s2 += v * v;
            yp[i] = v;
        }
        atomicAdd(&lsum[o], s);
        atomicAdd(&lsumsq[o], s2);
    }
    __syncthreads();
    for (int i = threadIdx.x; i < OUTC; i += blockDim.x) {
        atomicAdd(&gstats[i],        lsum[i]);
        atomicAdd(&gstats[OUTC + i], lsumsq[i]);
    }
}

// ---------------------------------------------------------------------------
// Kernel 3: finalize BN stats -> per-channel scale/shift
// ---------------------------------------------------------------------------
__global__ void stats_kernel(const float* __restrict__ gstats,
                             const float* __restrict__ gamma, const float* __restrict__ beta,
                             float* __restrict__ scale, float* __restrict__ shift) {
    int o = threadIdx.x;
    if (o < OUTC) {
        const float invN = 1.0f / (float)NPOS;
        float mean = gstats[o] * invN;
        float var  = gstats[OUTC + o] * invN - mean * mean;
        float sc   = gamma[o] * rsqrtf(var + EPSV);
        scale[o] = sc;
        shift[o] = beta[o] - mean * sc;
    }
}

// ---------------------------------------------------------------------------
// Kernel 4: BN apply + exact GELU (erf form)
// ---------------------------------------------------------------------------
__global__ void bn_gelu_kernel(const float* __restrict__ y,
                               const float* __restrict__ scale, const float* __restrict__ shift,
                               float* __restrict__ out) {
    const int total = BB * OUTC * HW;
    for (int i = blockIdx.x * blockDim.x + threadIdx.x; i < total; i += gridDim.x * blockDim.x) {
        int c = (i / HW) % OUTC;
        float v = y[i] * scale[c] + shift[c];
        out[i] = 0.5f * v * (1.0f + erff(v * 0.70710678118654752440f));
    }
}

// ---------------------------------------------------------------------------
// Launch
// ---------------------------------------------------------------------------
extern "C" void kernel_launch(void* const* d_in, const int* in_sizes, int n_in,
                              void* d_out, int out_size, void* d_ws, size_t ws_size,
                              hipStream_t stream) {
    const float* x     = (const float*)d_in[0];
    const float* cw    = (const float*)d_in[1];
    const float* cb    = (const float*)d_in[2];
    const float* gamma = (const float*)d_in[3];
    const float* beta  = (const float*)d_in[4];
    float* out = (float*)d_out;

    float* ws     = (float*)d_ws;
    float* xj     = ws;                                   // 19,267,584 f
    float* y      = xj + (size_t)BB * CC * HW;            // 38,535,168 f
    float* gstats = y + (size_t)BB * OUTC * HW;           // 384 f (sum | sumsq)
    float* scale  = gstats + 2 * OUTC;                    // 192 f
    float* shift  = scale + OUTC;                         // 192 f

    zero_stats_kernel<<<1, 512, 0, stream>>>(gstats);
    shiftmax_kernel<<<BB * CC, 256, 0, stream>>>(x, xj);
    gemm_wmma_kernel<<<NTILES / 8, 256, 0, stream>>>(x, xj, cw, cb, y, gstats);
    stats_kernel<<<1, 256, 0, stream>>>(gstats, gamma, beta, scale, shift);
    bn_gelu_kernel<<<4096, 256, 0, stream>>>(y, scale, shift, out);
}